// AnchorGate_18210661335096
// MI455X (gfx1250) — compile-verified
//
#include <hip/hip_runtime.h>
#include <math.h>
#include <stdint.h>

typedef __attribute__((ext_vector_type(2))) float v2f;
typedef __attribute__((ext_vector_type(8))) float v8f;

// ---------------------------------------------------------------------------
// Row squared-norm: out[r] = sum_d X[r][d]^2
// ---------------------------------------------------------------------------
__global__ __launch_bounds__(64)
void rownorm2_kernel(const float* __restrict__ X, float* __restrict__ out, int D) {
  const int r = blockIdx.x;
  const int t = threadIdx.x;
  float s = 0.f;
  for (int d = t; d < D; d += 64) { float x = X[(size_t)r * D + d]; s += x * x; }
  __shared__ float buf[64];
  buf[t] = s;
  __syncthreads();
  #pragma unroll
  for (int w = 32; w > 0; w >>= 1) {
    if (t < w) buf[t] += buf[t + w];
    __syncthreads();
  }
  if (t == 0) out[r] = buf[0];
}

// ---------------------------------------------------------------------------
// Async global->LDS copy of 64 bytes (4 x B128), ASYNCcnt-tracked.
// INST_OFFSET applies to both the LDS and the memory address (ISA 10.x),
// so one LDS base + one global base covers all four chunks.
// ---------------------------------------------------------------------------
__device__ __forceinline__ void async_copy64(uint32_t lds_addr, const float* gaddr) {
  asm volatile(
      "global_load_async_to_lds_b128 %0, %1, off\n\t"
      "global_load_async_to_lds_b128 %0, %1, off offset:16\n\t"
      "global_load_async_to_lds_b128 %0, %1, off offset:32\n\t"
      "global_load_async_to_lds_b128 %0, %1, off offset:48"
      :: "v"(lds_addr), "v"(gaddr) : "memory");
}

__device__ __forceinline__ void wait_async_0() {
  asm volatile("s_wait_asynccnt 0x0" ::: "memory");
}

// ---------------------------------------------------------------------------
// C[M x N] = X (M x D) * Y (N x D)^T   using V_WMMA_F32_16X16X4_F32.
//
// Block = 128 threads (2x2 waves), block tile 64x64, each wave a 32x32
// sub-tile (4 accumulators). K is consumed in chunks of 32, double-buffered
// in LDS via GLOBAL_LOAD_ASYNC_TO_LDS_B128 (chunk c+1 copy overlaps compute
// on chunk c; s_wait_asynccnt + one barrier per chunk).
//
// fp32 A 16x4 layout: lane L<16 -> {A[L][k], A[L][k+1]}, lane L+16 -> {A[L][k+2], A[L][k+3]}
// fp32 B 4x16 layout: lane L<16 -> {B[k][L], B[k+1][L]}, lane L+16 -> {B[k+2][L], B[k+3][L]}
// Since B[k][n] = Y[n][k], both fragments load identically (row-major 2-float chunks).
// ---------------------------------------------------------------------------
#define LDS_STRIDE 34   // 32 floats data + 2 pad: 8B-aligned rows, conflict-free b64 reads

__global__ __launch_bounds__(128)
void wmma_gemm_nt(const float* __restrict__ X, const float* __restrict__ Y,
                  float* __restrict__ C, int M, int N, int D) {
  __shared__ float Xs[2][64 * LDS_STRIDE];
  __shared__ float Ys[2][64 * LDS_STRIDE];

  const int tid  = threadIdx.x;
  const int lane = tid & 31;
  const int wave = tid >> 5;
  const int ntn  = N >> 6;
  const int tm   = (blockIdx.x / ntn) << 6;
  const int tn   = (blockIdx.x % ntn) << 6;
  const int wm   = (wave >> 1) << 5;          // wave's 32x32 sub-tile in block tile
  const int wn   = (wave & 1) << 5;

  // staging assignment: thread t copies 16 floats of row (t>>1), half (t&1)
  const int r  = tid >> 1;
  const int f0 = (tid & 1) << 4;
  const float* gx = X + (size_t)(tm + r) * D + f0;
  const float* gy = Y + (size_t)(tn + r) * D + f0;
  const uint32_t lx[2] = { (uint32_t)(uintptr_t)&Xs[0][r * LDS_STRIDE + f0],
                           (uint32_t)(uintptr_t)&Xs[1][r * LDS_STRIDE + f0] };
  const uint32_t ly[2] = { (uint32_t)(uintptr_t)&Ys[0][r * LDS_STRIDE + f0],
                           (uint32_t)(uintptr_t)&Ys[1][r * LDS_STRIDE + f0] };

  // fragment addressing
  const int row = lane & 15;
  const int kh  = (lane >> 4) << 1;           // 0 or 2

  v8f acc00 = {}, acc01 = {}, acc10 = {}, acc11 = {};

  const int nchunks = D >> 5;                 // K chunks of 32

  // prologue: stage chunk 0 into buffer 0
  async_copy64(lx[0], gx);
  async_copy64(ly[0], gy);
  wait_async_0();
  __syncthreads();

  for (int c = 0; c < nchunks; ++c) {
    const int cur = c & 1;
    const int nxt = cur ^ 1;
    if (c + 1 < nchunks) {                    // overlap next copy with compute
      const int kc = (c + 1) << 5;
      async_copy64(lx[nxt], gx + kc);
      async_copy64(ly[nxt], gy + kc);
    }

    const float* xa0 = &Xs[cur][(wm + row) * LDS_STRIDE + kh];
    const float* xa1 = xa0 + 16 * LDS_STRIDE;
    const float* yb0 = &Ys[cur][(wn + row) * LDS_STRIDE + kh];
    const float* yb1 = yb0 + 16 * LDS_STRIDE;

    #pragma unroll
    for (int kk = 0; kk < 32; kk += 4) {
      v2f a0 = *(const v2f*)(xa0 + kk);
      v2f a1 = *(const v2f*)(xa1 + kk);
      v2f b0 = *(const v2f*)(yb0 + kk);
      v2f b1 = *(const v2f*)(yb1 + kk);
      acc00 = __builtin_amdgcn_wmma_f32_16x16x4_f32(false, a0, false, b0, (short)0, acc00, false, false);
      acc01 = __builtin_amdgcn_wmma_f32_16x16x4_f32(false, a0, false, b1, (short)0, acc01, false, false);
      acc10 = __builtin_amdgcn_wmma_f32_16x16x4_f32(false, a1, false, b0, (short)0, acc10, false, false);
      acc11 = __builtin_amdgcn_wmma_f32_16x16x4_f32(false, a1, false, b1, (short)0, acc11, false, false);
    }

    if (c + 1 < nchunks) wait_async_0();      // next chunk landed
    __syncthreads();                          // all waves done reading cur / nxt visible
  }

  // C/D 16x16 fp32 layout: VGPR v -> row v (lanes 0-15) / row v+8 (lanes 16-31)
  const int cm = (lane >> 4) << 3;            // 0 or 8
  const int cn = lane & 15;
  float* c0 = C + (size_t)(tm + wm + cm) * N + (tn + wn) + cn;
  #pragma unroll
  for (int v = 0; v < 8; ++v) {
    c0[(size_t)v * N]              = acc00[v];
    c0[(size_t)v * N + 16]         = acc01[v];
    c0[((size_t)v + 16) * N]       = acc10[v];
    c0[((size_t)v + 16) * N + 16]  = acc11[v];
  }
}

// ---------------------------------------------------------------------------
// kNN over anchors (3 nearest, Euclidean, self excluded, ties -> lower index),
// then simplex vertex list and 4x4 squared-distance matrix per anchor.
// ---------------------------------------------------------------------------
__global__ __launch_bounds__(256)
void knn_kernel(const float* __restrict__ Gaa, const float* __restrict__ an,
                int* __restrict__ verts, float* __restrict__ d2v, int A) {
  const int a   = blockIdx.x;
  const int tid = threadIdx.x;

  float bd0 = INFINITY, bd1 = INFINITY, bd2 = INFINITY;
  int   bi0 = 0x7fffffff, bi1 = 0x7fffffff, bi2 = 0x7fffffff;
  const float ana = an[a];
  for (int j = tid; j < A; j += 256) {
    float d2 = ana + an[j] - 2.f * Gaa[(size_t)a * A + j];
    float d  = sqrtf(fmaxf(d2, 0.f));
    if (j == a) d = INFINITY;
    if (d < bd0)      { bd2 = bd1; bi2 = bi1; bd1 = bd0; bi1 = bi0; bd0 = d; bi0 = j; }
    else if (d < bd1) { bd2 = bd1; bi2 = bi1; bd1 = d;  bi1 = j; }
    else if (d < bd2) { bd2 = d;  bi2 = j; }
  }

  __shared__ float sd[768];
  __shared__ int   si[768];
  __shared__ int   vsel[4];
  sd[tid]       = bd0; si[tid]       = bi0;
  sd[256 + tid] = bd1; si[256 + tid] = bi1;
  sd[512 + tid] = bd2; si[512 + tid] = bi2;
  __syncthreads();

  if (tid == 0) {
    int s0 = -1, s1 = -1, s2 = -1;
    for (int t = 0; t < 3; ++t) {
      float best = INFINITY; int bidx = 0x7fffffff;
      for (int e = 0; e < 768; ++e) {
        int   idx = si[e];
        float d   = sd[e];
        if (idx == s0 || idx == s1 || idx == s2) continue;
        if (d < best || (d == best && idx < bidx)) { best = d; bidx = idx; }
      }
      if (t == 0) s0 = bidx; else if (t == 1) s1 = bidx; else s2 = bidx;
    }
    vsel[0] = a; vsel[1] = s0; vsel[2] = s1; vsel[3] = s2;
    verts[a * 4 + 0] = a;  verts[a * 4 + 1] = s0;
    verts[a * 4 + 2] = s1; verts[a * 4 + 3] = s2;
  }
  __syncthreads();

  if (tid < 16) {
    int i = tid >> 2, j = tid & 3;
    int vi = vsel[i], vj = vsel[j];
    d2v[(size_t)a * 16 + tid] = an[vi] + an[vj] - 2.f * Gaa[(size_t)vi * A + vj];
  }
}

// ---------------------------------------------------------------------------
// Per-(b,a) 6x6 Cayley-Menger determinant -> validity = sign * log(|det|+1e-12)
// Fully-unrolled Gaussian elimination with branchless partial pivoting so the
// matrix stays in VGPRs (no scratch).
// ---------------------------------------------------------------------------
__global__ __launch_bounds__(256)
void cm_det_kernel(const float* __restrict__ Gea, const float* __restrict__ en,
                   const float* __restrict__ an, const int* __restrict__ verts,
                   const float* __restrict__ d2v, float* __restrict__ validity, int A) {
  const int a = blockIdx.x * 256 + threadIdx.x;
  const int b = blockIdx.y;

  const float enb = en[b];
  float d2e[4];
  #pragma unroll
  for (int j = 0; j < 4; ++j) {
    int vj = verts[a * 4 + j];
    d2e[j] = enb + an[vj] - 2.f * Gea[(size_t)b * A + vj];
  }

  float m[6][6];
  m[0][0] = 0.f;
  #pragma unroll
  for (int j = 1; j < 6; ++j) { m[0][j] = 1.f; m[j][0] = 1.f; }
  m[1][1] = 0.f;
  #pragma unroll
  for (int j = 0; j < 4; ++j) { m[1][2 + j] = d2e[j]; m[2 + j][1] = d2e[j]; }
  #pragma unroll
  for (int i = 0; i < 4; ++i)
    #pragma unroll
    for (int j = 0; j < 4; ++j)
      m[2 + i][2 + j] = d2v[(size_t)a * 16 + i * 4 + j];

  float det = 1.f;
  #pragma unroll
  for (int c = 0; c < 6; ++c) {
    float best = fabsf(m[c][c]); int bp = c;
    #pragma unroll
    for (int r = c + 1; r < 6; ++r) {
      float v = fabsf(m[r][c]);
      if (v > best) { best = v; bp = r; }
    }
    #pragma unroll
    for (int r = c + 1; r < 6; ++r) {
      bool sw = (r == bp);
      #pragma unroll
      for (int cc = c; cc < 6; ++cc) {
        float t0 = m[c][cc], t1 = m[r][cc];
        m[c][cc] = sw ? t1 : t0;
        m[r][cc] = sw ? t0 : t1;
      }
      det = sw ? -det : det;
    }
    float piv = m[c][c];
    det *= piv;
    float inv = (piv != 0.f) ? 1.f / piv : 0.f;
    #pragma unroll
    for (int r = c + 1; r < 6; ++r) {
      float f = m[r][c] * inv;
      #pragma unroll
      for (int cc = c + 1; cc < 6; ++cc)
        m[r][cc] -= f * m[c][cc];
    }
  }

  float rd  = -det;                       // sign_corr = (-1)^(k+1), k=4
  float sgn = (rd > 0.f) ? 1.f : ((rd < 0.f) ? -1.f : 0.f);
  validity[(size_t)b * A + a] = sgn * logf(fabsf(rd) + 1e-12f);
}

// ---------------------------------------------------------------------------
// Fused per-row: normalize validity (mean/std ddof=1), ranks of (1 - cos) by
// counting in LDS, gate MLP (3->16 exact GELU ->1) + sigmoid.
// One block of 1024 threads per embedding row (A == 1024).
// ---------------------------------------------------------------------------
__global__ __launch_bounds__(1024)
void gate_kernel(const float* __restrict__ validity, const float* __restrict__ Gea,
                 const float* __restrict__ w1, const float* __restrict__ b1,
                 const float* __restrict__ w2, const float* __restrict__ b2,
                 float* __restrict__ out, int A) {
  const int b   = blockIdx.x;
  const int tid = threadIdx.x;

  __shared__ float sdist[1024];
  __shared__ float rbuf[1024];
  __shared__ float sw1[48], sb1[16], sw2[16];
  __shared__ float sb2, smean, sstd;

  float v    = validity[(size_t)b * A + tid];
  float cosv = Gea[(size_t)b * A + tid];
  sdist[tid] = 1.f - cosv;
  if (tid < 48) sw1[tid] = w1[tid];
  if (tid < 16) { sb1[tid] = b1[tid]; sw2[tid] = w2[tid]; }
  if (tid == 0) sb2 = b2[0];

  rbuf[tid] = v;
  __syncthreads();
  #pragma unroll
  for (int s = 512; s > 0; s >>= 1) {
    if (tid < s) rbuf[tid] += rbuf[tid + s];
    __syncthreads();
  }
  if (tid == 0) smean = rbuf[0] * (1.f / 1024.f);
  __syncthreads();
  float mean = smean;
  float dv   = v - mean;
  rbuf[tid]  = dv * dv;
  __syncthreads();
  #pragma unroll
  for (int s = 512; s > 0; s >>= 1) {
    if (tid < s) rbuf[tid] += rbuf[tid + s];
    __syncthreads();
  }
  if (tid == 0) sstd = fmaxf(sqrtf(rbuf[0] * (1.f / 1023.f)), 1e-8f);
  __syncthreads();

  float vnorm = dv / sstd;

  // rank = #elements strictly smaller (stable: ties broken by lower index)
  float myd = sdist[tid];
  int cnt = 0;
  #pragma unroll 4
  for (int k = 0; k < 1024; ++k) {
    float dk = sdist[k];
    cnt += (dk < myd || (dk == myd && k < tid)) ? 1 : 0;
  }
  float rank = (float)cnt * (1.f / 1023.f);

  float g = sb2;
  #pragma unroll
  for (int j = 0; j < 16; ++j) {
    float h  = vnorm * sw1[j] + cosv * sw1[16 + j] + rank * sw1[32 + j] + sb1[j];
    float ge = 0.5f * h * (1.f + erff(h * 0.70710678118654752f));
    g += ge * sw2[j];
  }
  out[(size_t)b * A + tid] = 1.f / (1.f + expf(-g));
}

// ---------------------------------------------------------------------------
extern "C" void kernel_launch(void* const* d_in, const int* in_sizes, int n_in,
                              void* d_out, int out_size, void* d_ws, size_t ws_size,
                              hipStream_t stream) {
  const int B = 2048, A = 1024, D = 512;
  const float* emb = (const float*)d_in[0];
  const float* anc = (const float*)d_in[1];
  // d_in[2] = tri (unused by the cm_gate path)
  const float* w1  = (const float*)d_in[3];
  const float* b1  = (const float*)d_in[4];
  const float* w2  = (const float*)d_in[5];
  const float* b2  = (const float*)d_in[6];
  float* out = (float*)d_out;

  float* ws       = (float*)d_ws;
  float* Gaa      = ws;  ws += (size_t)A * A;   // 4 MB
  float* Gea      = ws;  ws += (size_t)B * A;   // 8 MB
  float* validity = ws;  ws += (size_t)B * A;   // 8 MB
  float* en       = ws;  ws += B;
  float* an       = ws;  ws += A;
  float* d2v      = ws;  ws += (size_t)A * 16;
  int*   verts    = (int*)ws;                   // A*4 ints

  rownorm2_kernel<<<B, 64, 0, stream>>>(emb, en, D);
  rownorm2_kernel<<<A, 64, 0, stream>>>(anc, an, D);

  // Gaa = anchors @ anchors^T : 64x64 block tiles -> 16*16 = 256 blocks
  wmma_gemm_nt<<<(A / 64) * (A / 64), 128, 0, stream>>>(anc, anc, Gaa, A, A, D);
  // Gea = embedding @ anchors^T : 32*16 = 512 blocks
  wmma_gemm_nt<<<(B / 64) * (A / 64), 128, 0, stream>>>(emb, anc, Gea, B, A, D);

  knn_kernel<<<A, 256, 0, stream>>>(Gaa, an, verts, d2v, A);

  cm_det_kernel<<<dim3(A / 256, B), 256, 0, stream>>>(Gea, en, an, verts, d2v, validity, A);

  gate_kernel<<<B, 1024, 0, stream>>>(validity, Gea, w1, b1, w2, b2, out, A);
}